// SemanticMaskVectorQuantizer_6811818131703
// MI455X (gfx1250) — compile-verified
//
#include <hip/hip_runtime.h>
#include <hip/hip_bf16.h>

// ---------------------------------------------------------------------------
// VQ-VAE vector quantizer for MI455X (gfx1250, wave32).
//   z_e:      [16, 256, 32, 32] f32   (N = 16*32*32 = 16384 rows, D = 256)
//   codebook: [8192, 256] f32          (K = 8192)
// argmin_k ( ||e_k||^2 - 2 z.e_k )  == fused GEMM + argmin via
// V_WMMA_F32_16X16X4_F32 (exact fp32; bf16 would risk flipping argmin since
// top-2 distance gaps are ~1e-5). A = -2*z staged in LDS; accumulators start
// at ||e_k||^2 so the WMMA accumulator directly holds the distance.
//
// This round: double-buffered codebook tiles filled with
// GLOBAL_LOAD_ASYNC_TO_LDS_B128 (ASYNCcnt) so the 32 KB/group staging is
// hidden behind the 128-WMMA compute phase. LDS = 64 KB z + 2x32 KB B.
// ---------------------------------------------------------------------------

typedef __attribute__((ext_vector_type(2))) float v2f;
typedef __attribute__((ext_vector_type(4))) float v4f;
typedef __attribute__((ext_vector_type(8))) float v8f;

#define K_CODES 8192
#define D_DIM   256
#define N_VEC   16384
#define HW_SZ   1024

#define OUT_ZQST   0
#define OUT_ZQ     4194304
#define OUT_CODES  8388608
#define OUT_SCAL   8404992   // 6 scalars follow

#define ASYNC_WAIT0() asm volatile("s_wait_asynccnt 0" ::: "memory")

// ---------------------------------------------------------------- e2[k] = ||e_k||^2
__global__ void vq_e2_kernel(const float* __restrict__ cb, float* __restrict__ e2) {
    const int lane = threadIdx.x & 31;
    const int wid  = threadIdx.x >> 5;
    const int row  = blockIdx.x * 8 + wid;          // grid = 1024, block = 256
    const float* r = cb + row * D_DIM;
    float s = 0.0f;
#pragma unroll
    for (int i = 0; i < 8; ++i) { float v = r[lane + 32 * i]; s += v * v; }
#pragma unroll
    for (int m = 16; m >= 1; m >>= 1) s += __shfl_xor(s, m, 32);
    if (lane == 0) e2[row] = s;
}

// ---- async-stage one 32-code (32 KB) tile: global -> LDS, XOR-swizzled.
// 2048 x b128 transfers; 128 threads issue 16 each (fire-and-forget, ASYNCcnt).
__device__ __forceinline__ void async_stage_tile(const float* __restrict__ cb_tile,
                                                 float* __restrict__ dst, int tid) {
#pragma unroll
    for (int it = 0; it < 16; ++it) {
        int idx = tid + it * 128;                 // 0..2047
        int c = idx >> 6;                         // code row 0..31
        int f = idx & 63;                         // float4 within row
        unsigned lds_off =
            (unsigned)(unsigned long long)(uintptr_t)
                &dst[c * 256 + ((f << 2) ^ ((c & 15) << 2))];
        int goff = (c * 256 + (f << 2)) * 4;      // byte offset within tile
        asm volatile("global_load_async_to_lds_b128 %0, %1, %2"
                     :: "v"(lds_off), "v"(goff), "s"(cb_tile)
                     : "memory");
    }
}

// ---------------------------------------------------------------- main: GEMM + argmin
// block = 128 threads (4 waves); each wave owns 16 z-rows; ping-pong B tiles.
__global__ void __launch_bounds__(128)
vq_argmin_kernel(const float* __restrict__ z_e, const float* __restrict__ cb,
                 const float* __restrict__ e2g, int* __restrict__ codes_i,
                 float* __restrict__ codes_f) {
    __shared__ __align__(16) float zt[64 * 256];        // -2*z, XOR-swizzled (64 KB)
    __shared__ __align__(16) float bt[2 * 32 * 256];    // double-buffered B (64 KB)

    const int tid    = threadIdx.x;
    const int lane   = tid & 31;
    const int waveId = tid >> 5;
    const int row0   = blockIdx.x * 64;            // global n base (grid = 256)
    const int bimg   = row0 >> 10;                 // batch index (64 rows never cross b)
    const int hw0    = row0 & 1023;

    // kick off async fill of B tile 0 first; it runs while we stage z below
    async_stage_tile(cb, bt, tid);

    // ---- stage z tile: rows [row0, row0+64), dims 0..255, scaled by -2.
    // z_e element (n, d) lives at z_e[(b*256 + d)*1024 + hw]; coalesce along hw.
    {
        const int rsub = tid & 63;       // row within block
        const int dgrp = tid >> 6;       // 0..1
        const int hw   = hw0 + rsub;
        const int sw   = (rsub & 15) << 2;
        for (int d0 = 0; d0 < D_DIM; d0 += 2) {
            int d = d0 + dgrp;
            float v = z_e[(bimg * D_DIM + d) * HW_SZ + hw];
            zt[rsub * 256 + (d ^ sw)] = -2.0f * v;
        }
    }
    ASYNC_WAIT0();
    __syncthreads();

    const int mrow = waveId * 16 + (lane & 15);    // local A row for this lane
    const int swz  = (lane & 15) << 2;             // XOR swizzle for both A and B rows
    const int hi2  = (lane >> 4) << 1;             // 0 (lanes 0-15) or 2 (lanes 16-31)
    const float* zrow = zt + mrow * 256;

    float runmin[8];
    int   runidx[8];
#pragma unroll
    for (int r = 0; r < 8; ++r) { runmin[r] = 3.0e38f; runidx[r] = 0; }

    for (int g = 0; g < K_CODES / 32; ++g) {
        const int code0 = g * 32;
        const int cur   = g & 1;
        const float* btc = bt + cur * (32 * 256);

        // fire async fill of the NEXT tile into the other buffer (hidden by WMMAs)
        if (g + 1 < K_CODES / 32)
            async_stage_tile(cb + (code0 + 32) * D_DIM, bt + (cur ^ 1) * (32 * 256), tid);

        // accumulators start at ||e_k||^2  (k depends only on lane&15 + sub-tile)
        float ev0 = e2g[code0 + (lane & 15)];
        float ev1 = e2g[code0 + 16 + (lane & 15)];
        v8f acc0 = { ev0, ev0, ev0, ev0, ev0, ev0, ev0, ev0 };
        v8f acc1 = { ev1, ev1, ev1, ev1, ev1, ev1, ev1, ev1 };

        const float* b0row = btc + (lane & 15) * 256;
        const float* b1row = b0row + 16 * 256;

#pragma unroll 8
        for (int d0 = 0; d0 < D_DIM; d0 += 4) {
            int koff = (d0 + hi2) ^ swz;            // conflict-free b64 LDS reads
            v2f a  = *(const v2f*)(zrow  + koff);
            v2f b0 = *(const v2f*)(b0row + koff);
            v2f b1 = *(const v2f*)(b1row + koff);
            acc0 = __builtin_amdgcn_wmma_f32_16x16x4_f32(false, a, false, b0,
                                                         (short)0, acc0, false, false);
            acc1 = __builtin_amdgcn_wmma_f32_16x16x4_f32(false, a, false, b1,
                                                         (short)0, acc1, false, false);
        }

        // ---- running argmin; candidate codes strictly increase -> first-min kept
        int c0 = code0 + (lane & 15);
        int c1 = c0 + 16;
#pragma unroll
        for (int r = 0; r < 8; ++r) {
            float d0v = acc0[r];
            if (d0v < runmin[r]) { runmin[r] = d0v; runidx[r] = c0; }
            float d1v = acc1[r];
            if (d1v < runmin[r]) { runmin[r] = d1v; runidx[r] = c1; }
        }

        // drain this wave's async fills of the next buffer, then block-sync
        ASYNC_WAIT0();
        __syncthreads();
    }

    // ---- reduce across the 16 code-columns in each half-wave (ties -> lower idx)
#pragma unroll
    for (int r = 0; r < 8; ++r) {
        float v = runmin[r];
        int  ix = runidx[r];
#pragma unroll
        for (int m = 1; m <= 8; m <<= 1) {
            float ov = __shfl_xor(v, m, 32);
            int   oi = __shfl_xor(ix, m, 32);
            if (ov < v || (ov == v && oi < ix)) { v = ov; ix = oi; }
        }
        if ((lane & 15) == 0) {
            // acc VGPR r: lanes 0-15 -> M=r, lanes 16-31 -> M=r+8
            int mg = row0 + waveId * 16 + r + ((lane >> 4) << 3);
            codes_i[mg] = ix;
            codes_f[mg] = (float)ix;
        }
    }
}

// ---------------------------------------------------------------- gather + loss partials
__global__ void vq_gather_kernel(const float* __restrict__ z_e,
                                 const float* __restrict__ cb,
                                 const int* __restrict__ codes_i,
                                 float* __restrict__ out,
                                 float* __restrict__ partials) {
    __shared__ float sred[256];
    const int o  = blockIdx.x * 256 + threadIdx.x;       // grid = 16384
    const int b  = o >> 18;                               // o = b*262144 + d*1024 + hw
    const int d  = (o >> 10) & 255;
    const int hw = o & 1023;
    const int n  = b * HW_SZ + hw;
    const int code = codes_i[n];
    float q    = cb[code * D_DIM + d];
    float ze   = z_e[o];
    float diff = q - ze;
    out[OUT_ZQST + o] = ze + diff;    // matches z_e + (z_q - z_e) rounding
    out[OUT_ZQ   + o] = q;
    sred[threadIdx.x] = diff * diff;
    __syncthreads();
    for (int s = 128; s > 0; s >>= 1) {
        if (threadIdx.x < s) sred[threadIdx.x] += sred[threadIdx.x + s];
        __syncthreads();
    }
    if (threadIdx.x == 0) partials[blockIdx.x] = sred[0];
}

// ---------------------------------------------------------------- histogram
__global__ void vq_zero_counts(int* __restrict__ counts) {   // grid = 32
    counts[blockIdx.x * 256 + threadIdx.x] = 0;
}
__global__ void vq_hist(const int* __restrict__ codes_i, int* __restrict__ counts) {
    int n = blockIdx.x * 256 + threadIdx.x;                  // grid = 64
    atomicAdd(&counts[codes_i[n]], 1);                       // integer: deterministic
}

// ---------------------------------------------------------------- scalars (single block)
__global__ void vq_finalize(const float* __restrict__ partials,
                            const int* __restrict__ counts,
                            float* __restrict__ out) {
    __shared__ float sA[256];
    __shared__ float sB[256];
    __shared__ int   sU[256];
    const int t = threadIdx.x;
    float sum = 0.0f;
    for (int i = t; i < 16384; i += 256) sum += partials[i];   // fixed order
    float ent = 0.0f;
    int used = 0;
    for (int i = t; i < K_CODES; i += 256) {
        int c = counts[i];
        if (c > 0) {
            used += 1;
            float p = (float)c * (1.0f / 16384.0f);
            ent += p * logf(fmaxf(p, 1e-10f));
        }
    }
    sA[t] = sum; sB[t] = ent; sU[t] = used;
    __syncthreads();
    for (int s = 128; s > 0; s >>= 1) {
        if (t < s) { sA[t] += sA[t + s]; sB[t] += sB[t + s]; sU[t] += sU[t + s]; }
        __syncthreads();
    }
    if (t == 0) {
        float loss = sA[0] * (1.0f / 4194304.0f);
        out[OUT_SCAL + 0] = loss;               // codebook_loss
        out[OUT_SCAL + 1] = loss;               // commitment_loss (same value)
        out[OUT_SCAL + 2] = expf(-sB[0]);       // perplexity
        float uf = (float)sU[0] * (1.0f / 8192.0f);
        out[OUT_SCAL + 3] = (float)sU[0];       // used_code_count
        out[OUT_SCAL + 4] = uf;                 // used_code_fraction
        out[OUT_SCAL + 5] = 1.0f - uf;          // dead_code_fraction
    }
}

// ---------------------------------------------------------------- launcher
extern "C" void kernel_launch(void* const* d_in, const int* in_sizes, int n_in,
                              void* d_out, int out_size, void* d_ws, size_t ws_size,
                              hipStream_t stream) {
    const float* z_e = (const float*)d_in[0];   // [16,256,32,32]
    const float* cb  = (const float*)d_in[1];   // [8192,256]
    float* out = (float*)d_out;

    char* ws = (char*)d_ws;
    float* e2       = (float*)(ws);                          // 8192 f32   (32 KB)
    int*   codes_i  = (int*)  (ws + 32768);                  // 16384 i32  (64 KB)
    int*   counts   = (int*)  (ws + 32768 + 65536);          // 8192 i32   (32 KB)
    float* partials = (float*)(ws + 32768 + 65536 + 32768);  // 16384 f32  (64 KB)

    vq_e2_kernel    <<<1024, 256, 0, stream>>>(cb, e2);
    vq_argmin_kernel<<< 256, 128, 0, stream>>>(z_e, cb, e2, codes_i, out + OUT_CODES);
    vq_gather_kernel<<<16384, 256, 0, stream>>>(z_e, cb, codes_i, out, partials);
    vq_zero_counts  <<<  32, 256, 0, stream>>>(counts);
    vq_hist         <<<  64, 256, 0, stream>>>(codes_i, counts);
    vq_finalize     <<<   1, 256, 0, stream>>>(partials, counts, out);
}